// BBPMAssociativeModel_20830591386040
// MI455X (gfx1250) — compile-verified
//
#include <hip/hip_runtime.h>

// Problem: B=32, T=4096, D=512 (derived from in_sizes at launch).
// ws layout (4-byte units):
//   [0..31]   qh   (int)   query hashes
//   [32..63]  idx  (int)   argmax flat write index per query, -1 = none
//   [64..)    R    (float) gathered retrieved rows, 32 x D, 256B-aligned

typedef __attribute__((ext_vector_type(2))) float v2f;
typedef __attribute__((ext_vector_type(4))) float v4f;
typedef __attribute__((ext_vector_type(8))) float v8f;

// ---- wave32 row hash: trunc(sum(row * 1000)) as int32 --------------------
// NT loads: x is a 268 MB one-shot stream ( > 192 MB L2 ), don't cache it.
__device__ __forceinline__ int hash_row_wave32(const float* __restrict__ row, int lane) {
    float s = 0.0f;
    // 512 floats = 128 xfloat4; 32 lanes x 4 iters, fully coalesced b128 loads
    const v4f* r4 = (const v4f*)row;
#pragma unroll
    for (int it = 0; it < 4; ++it) {
        v4f v = __builtin_nontemporal_load(r4 + it * 32 + lane);
        s += v.x * 1000.0f + v.y * 1000.0f + v.z * 1000.0f + v.w * 1000.0f;
    }
    // butterfly reduction -> every lane holds the full sum (wave32!)
#pragma unroll
    for (int off = 16; off >= 1; off >>= 1)
        s += __shfl_xor(s, off, 32);
    return (int)truncf(s);
}

// ---- kernel 1: query hashes + idx init (1 block, 32 waves) ---------------
__global__ __launch_bounds__(1024) void qhash_init_kernel(
    const float* __restrict__ x, int* __restrict__ qh, int* __restrict__ idx,
    int T, int D) {
    const int b    = threadIdx.x >> 5;   // 0..31, one wave per query row
    const int lane = threadIdx.x & 31;
    const float* row = x + ((size_t)b * T + (T - 1)) * (size_t)D;
    int h = hash_row_wave32(row, lane);
    if (lane == 0) {
        qh[b]  = h;
        idx[b] = -1;
    }
}

// ---- kernel 2: hash each write row, match vs all 32 queries --------------
// one wave per write row n (flat order n = t*32 + b); lane L holds qh[L]
__global__ __launch_bounds__(256) void whash_match_kernel(
    const float* __restrict__ x, const int* __restrict__ qh,
    int* __restrict__ idx, int T, int D, int N) {
    const int gw   = blockIdx.x * (blockDim.x >> 5) + (threadIdx.x >> 5);
    const int lane = threadIdx.x & 31;
    if (gw >= N) return;
    const int t = gw >> 5;      // n / B  (B == 32)
    const int b = gw & 31;      // n % B
    const float* row = x + ((size_t)b * T + t) * (size_t)D;
    int h = hash_row_wave32(row, lane);   // all lanes hold the row hash
    int q = qh[lane];                     // lane L owns query L
    if (h == q) atomicMax(&idx[lane], gw);   // "last write wins" == max n
}

// ---- kernel 3: gather retrieved rows (zeros if no match) -----------------
__global__ __launch_bounds__(128) void gather_kernel(
    const float* __restrict__ x, const int* __restrict__ idx,
    float* __restrict__ R, int T, int D) {
    const int b = blockIdx.x;           // 0..31
    const int i = idx[b];
    v4f v = {0.f, 0.f, 0.f, 0.f};
    if (i >= 0) {
        const int t  = i >> 5;
        const int bi = i & 31;
        v = ((const v4f*)(x + ((size_t)bi * T + t) * (size_t)D))[threadIdx.x];
    }
    ((v4f*)(R + (size_t)b * D))[threadIdx.x] = v;  // 128 thr x 16B = 512 floats
}

// ---- kernel 4: out[32, D] = R[32, D] @ W^T + bias via V_WMMA_F32_16X16X4_F32
// grid.x = D/16 N-tiles; 2 waves per block = 2 M-tiles (M = 32)
__global__ __launch_bounds__(64) void gemm_wmma_f32_kernel(
    const float* __restrict__ R, const float* __restrict__ W,
    const float* __restrict__ bias, float* __restrict__ out, int D) {
    const int ntile = blockIdx.x;          // 16-wide column tile of out
    const int mwave = threadIdx.x >> 5;    // 0..1 -> rows [16m, 16m+16)
    const int lane  = threadIdx.x & 31;
    const int half  = lane >> 4;           // ISA: upper 16 lanes hold K+2/K+3 (A)
    const int l15   = lane & 15;           //      and row M+8 / K-row+2 (B,C,D)

    // A: row M = l15 of this wave's M-tile; fp32 pair (K+2*half, K+2*half+1)
    const float* Arow = R + (size_t)(mwave * 16 + l15) * D;
    // B[k][n] = W^T[k][n] = W[n][k] with n = ntile*16 + l15 -> row-major W row
    const float* Brow = W + (size_t)(ntile * 16 + l15) * D;

    v8f c = {0.f, 0.f, 0.f, 0.f, 0.f, 0.f, 0.f, 0.f};
    for (int k = 0; k < D; k += 4) {
        v2f a = *(const v2f*)(Arow + k + 2 * half);
        v2f b = *(const v2f*)(Brow + k + 2 * half);
        // D = A(16x4) x B(4x16) + C ; 8-arg f32/f16/bf16 pattern
        c = __builtin_amdgcn_wmma_f32_16x16x4_f32(
            /*neg_a=*/false, a, /*neg_b=*/false, b,
            /*c_mod=*/(short)0, c, /*reuse_a=*/false, /*reuse_b=*/false);
    }

    // C/D 16x16 layout: lane half selects M base (0 or 8), VGPR v = M offset
    const int n = ntile * 16 + l15;
    const float bn = bias[n];
#pragma unroll
    for (int v = 0; v < 8; ++v) {
        const int m = mwave * 16 + half * 8 + v;
        out[(size_t)m * D + n] = c[v] + bn;
    }
}

extern "C" void kernel_launch(void* const* d_in, const int* in_sizes, int n_in,
                              void* d_out, int out_size, void* d_ws, size_t ws_size,
                              hipStream_t stream) {
    const float* x    = (const float*)d_in[0];
    // d_in[1] = hx_list: present in the reference signature but unused by it
    const float* W    = (const float*)d_in[2];
    const float* bias = (const float*)d_in[3];
    float* out = (float*)d_out;

    const int B = in_sizes[1];            // 32
    const int D = in_sizes[3];            // 512
    const int T = in_sizes[0] / (B * D);  // 4096
    const int N = (T - 1) * B;            // 131040 write rows

    int*   qh  = (int*)d_ws;
    int*   idx = qh + 32;
    float* R   = (float*)((char*)d_ws + 256);   // 16B-aligned gather buffer

    // 1) query hashes + idx init (one wave per query)
    qhash_init_kernel<<<1, 32 * 32, 0, stream>>>(x, qh, idx, T, D);

    // 2) hash all write rows (streams 268 MB of x once, NT) + fused match
    const int wavesPerBlock = 256 / 32;
    const int blocks = (N + wavesPerBlock - 1) / wavesPerBlock;
    whash_match_kernel<<<blocks, 256, 0, stream>>>(x, qh, idx, T, D, N);

    // 3) gather retrieved[B, D]
    gather_kernel<<<B, D / 4, 0, stream>>>(x, idx, R, T, D);

    // 4) out = R @ W^T + bias via fp32 WMMA
    gemm_wmma_f32_kernel<<<D / 16, 64, 0, stream>>>(R, W, bias, out, D);
}